// AP_9139690405900
// MI455X (gfx1250) — compile-verified
//
#include <hip/hip_runtime.h>
#include <stdint.h>

// ---------------------------------------------------------------------------
// AP rank loss, CDNA5 (gfx1250, wave32).
// C = 16384 elements, 16 classes, scalar f32 output.
//
// Pipeline (all deterministic; only integer atomics used):
//  k_build    : column-major scores, sortable keys, label-masked u64 keys
//  k_count    : per-class positive counts (int atomics)
//  k_prefix   : exclusive prefix of counts
//  k_scatter  : compact positives per class (slot order irrelevant)
//  k_tbl      : per-class table  tbl[m] = (1/m)/P   (matches reference rounding)
//  k_rankpos  : counting-sort positives descending (ties -> smaller index)
//  k_negrank  : ascending rank of each negative by u64-key counting
//  k_scan_wmma: per-negative suffix-sum argmax; 16x16 delta chunks, suffix
//               sums via 4x V_WMMA_F32_16X16X4_F32 (D = Delta * LowerTri(1)),
//               branch-free masked A construction, per-row carry/argmax merge
//               with reference tie semantics (first max -> smallest r)
//  k_finalize : per-class cumsum/histogram combine + fixed-order reductions
//  k_total    : sum / 16
// ---------------------------------------------------------------------------

#define NCLS 16
#define CDIM 16384

typedef __attribute__((ext_vector_type(2))) float v2f;
typedef __attribute__((ext_vector_type(8))) float v8f;

// ---- workspace layout (bytes) ----
#define OFF_COLSC   (size_t)0                      // float [NCLS*CDIM]
#define OFF_KEY32   (size_t)(1048576)              // uint  [NCLS*CDIM]
#define OFF_NK      (size_t)(2097152)              // u64   [NCLS*CDIM]
#define OFF_NEGRANK (size_t)(4194304)              // int   [NCLS*CDIM]
#define OFF_TBL     (size_t)(5242880)              // float [NCLS*(CDIM+2)]
#define OFF_HIST    (size_t)(6291584)              // int   [NCLS*(CDIM+2)]
#define OFF_SMDOT   (size_t)(7340288)              // float [NCLS*CDIM]
#define OFF_SMSUM   (size_t)(8388864)              // float [NCLS*CDIM]
#define OFF_POSDK   (size_t)(9437440)              // u64   [CDIM]
#define OFF_POSVAL  (size_t)(9568512)              // float [CDIM]
#define OFF_SPSORT  (size_t)(9634048)              // float [CDIM]
#define OFF_PCNT    (size_t)(9699584)              // int   [16]
#define OFF_POFF    (size_t)(9699648)              // int   [16]
#define OFF_PFILL   (size_t)(9699712)              // int   [16]
#define OFF_CLOSS   (size_t)(9699776)              // float [16]
#define WS_TOTAL    (size_t)(9699840)

__device__ __forceinline__ unsigned sortkey(float v) {
  unsigned u = __float_as_uint(v);
  return (u & 0x80000000u) ? ~u : (u | 0x80000000u);  // ascending order key
}

__global__ void k_build(const float* __restrict__ sc, const int* __restrict__ lab,
                        float* __restrict__ colsc, unsigned* __restrict__ key32,
                        unsigned long long* __restrict__ nk) {
  int t = blockIdx.x * blockDim.x + threadIdx.x;  // [0, NCLS*CDIM)
  int c = t / CDIM, i = t - c * CDIM;
  float v = sc[i * NCLS + c];
  unsigned u = sortkey(v);
  colsc[t] = v;
  key32[t] = u;
  nk[t] = (lab[i] == c) ? ~0ULL : ((((unsigned long long)u) << 32) | (unsigned)i);
}

__global__ void k_count(const int* __restrict__ lab, int* __restrict__ pcnt) {
  int i = blockIdx.x * blockDim.x + threadIdx.x;
  if (i < CDIM) atomicAdd(&pcnt[lab[i] & 15], 1);
}

__global__ void k_prefix(const int* __restrict__ pcnt, int* __restrict__ poff) {
  if (blockIdx.x == 0 && threadIdx.x == 0) {
    int r = 0;
    for (int c = 0; c < NCLS; ++c) { poff[c] = r; r += pcnt[c]; }
  }
}

__global__ void k_scatter(const float* __restrict__ sc, const int* __restrict__ lab,
                          const unsigned* __restrict__ key32, const int* __restrict__ poff,
                          int* __restrict__ pfill, unsigned long long* __restrict__ posdk,
                          float* __restrict__ posval) {
  int i = blockIdx.x * blockDim.x + threadIdx.x;
  if (i >= CDIM) return;
  int c = lab[i] & 15;
  int slot = atomicAdd(&pfill[c], 1);
  int p = poff[c] + slot;
  unsigned u = key32[c * CDIM + i];
  // descending value, ties -> smaller original index (ascending u64 compare)
  posdk[p] = (((unsigned long long)(~u)) << 32) | (unsigned)i;
  posval[p] = sc[i * NCLS + c];
}

__global__ void k_tbl(const int* __restrict__ pcnt, float* __restrict__ tblc) {
  int t = blockIdx.x * blockDim.x + threadIdx.x;
  if (t >= NCLS * (CDIM + 2)) return;
  int c = t / (CDIM + 2), m = t - c * (CDIM + 2);
  int P = pcnt[c];
  if (P <= 0 || m < 2) return;
  float r = 1.0f / (float)m;     // matches reference: 1/(j+k) then /P
  tblc[t] = r / (float)P;
}

__global__ void k_rankpos(const int* __restrict__ pcnt, const int* __restrict__ poff,
                          const unsigned long long* __restrict__ posdk,
                          const float* __restrict__ posval, float* __restrict__ spsort) {
  int c = blockIdx.x;
  int P = pcnt[c], o = poff[c];
  for (int p = threadIdx.x; p < P; p += blockDim.x) {
    unsigned long long me = posdk[o + p];
    int r = 0;
    for (int q = 0; q < P; ++q) r += (posdk[o + q] < me) ? 1 : 0;
    spsort[o + r] = posval[o + p];
  }
}

__global__ void k_negrank(const unsigned long long* __restrict__ nk, int* __restrict__ negrank) {
  int c = blockIdx.y;
  int i = blockIdx.x * blockDim.x + threadIdx.x;
  const unsigned long long* col = nk + (size_t)c * CDIM;
  unsigned long long me = col[i];
  if (me == ~0ULL) { negrank[c * CDIM + i] = 0; return; }  // positive of class c
  int cnt = 0;
  #pragma unroll 4
  for (int m = 0; m < CDIM; ++m) cnt += (col[m] < me) ? 1 : 0;
  negrank[c * CDIM + i] = cnt + 1;  // 1-indexed ascending rank among negatives
}

// Main scan: each wave owns 16 element rows of one class column.
// Chunk of 16 k-values: suffix sums of the 16x16 delta block computed as
// D = A(16x4)*B(4x16) accumulated over 4 K-slices with v_wmma_f32_16x16x4_f32,
// where B is the lower-triangular ones matrix T[k,r] = (k >= r).
// A-elements are built branch-free: clamped in-bounds loads * exact 0/1 mask.
__global__ __launch_bounds__(256) void k_scan_wmma(
    const float* __restrict__ colsc, const int* __restrict__ negrank,
    const float* __restrict__ tblc, const float* __restrict__ spsort,
    const int* __restrict__ pcnt, const int* __restrict__ poff,
    int* __restrict__ hist, float* __restrict__ smdot, float* __restrict__ smsum) {
  __shared__ float xfer[8][256];  // per-wave 16x16 D transfer buffer
  const int c = blockIdx.y;
  const int P = pcnt[c];
  const int N = CDIM - P;
  if (P == 0 || N == 0) return;
  const float Pf = (float)P, Nf = (float)N;
  const float c2 = 2.0f / (Pf * Nf);
  const int lane = threadIdx.x & 31;
  const int wave = threadIdx.x >> 5;
  const int m = lane & 15;                      // matrix row owned by this lane (A) / col (B,D)
  const int row = blockIdx.x * 128 + wave * 16 + m;  // element index
  const int jr = negrank[c * CDIM + row];
  const bool valid = jr > 0;
  const float validf = valid ? 1.0f : 0.0f;
  const int j = valid ? jr : 1;                 // clamped: keeps tbl reads in-bounds
  const float v = colsc[c * CDIM + row];        // s_minus[j-1] for valid rows
  const float* tbl = tblc + c * (CDIM + 2);
  const float* sp = spsort + poff[c];
  __builtin_prefetch(sp, 0, 1);                 // global_prefetch_b8
  const int kloA = (lane < 16) ? 0 : 2;         // K-local base held by this lane
  float carry = 0.0f, best = 0.0f;              // suffix(P+1) == 0 is the initial best
  int bestr = P + 1;
  const int nch = (P + 15) >> 4;
  for (int ch = nch - 1; ch >= 0; --ch) {       // right-to-left over k chunks
    const int kc = ch << 4;                     // chunk covers k = kc+1 .. kc+16 (1-indexed)
    v8f D = {0, 0, 0, 0, 0, 0, 0, 0};
    #pragma unroll
    for (int s = 0; s < 4; ++s) {
      const int kl0 = 4 * s + kloA;             // k-local in [0,15]
      const int k0 = kc + kl0 + 1;              // 1-indexed global k
      const int k1 = k0 + 1;
      // branch-free: clamp indices in-bounds, mask pad/invalid lanes to exact 0
      const int k0c = (k0 <= P) ? k0 : P;
      const int k1c = (k1 <= P) ? k1 : P;
      const float m0 = (k0 <= P) ? validf : 0.0f;
      const float m1 = (k1 <= P) ? validf : 0.0f;
      const float a0 = m0 * (tbl[j + k0c] - (sp[k0c - 1] - v) * c2);
      const float a1 = m1 * (tbl[j + k1c] - (sp[k1c - 1] - v) * c2);
      v2f A = {a0, a1};                         // A: 16x4 f32 layout (M=lane&15, K in VGPR/half)
      v2f B = {(kl0 >= m) ? 1.0f : 0.0f,        // B: T[k,r]=(k>=r), N=lane&15, K mirrors A
               (kl0 + 1 >= m) ? 1.0f : 0.0f};
      D = __builtin_amdgcn_wmma_f32_16x16x4_f32(false, A, false, B, (short)0, D, false, false);
    }
    // D layout: (VGPR n, lane L) = D[M = n + 8*(L>>4)][r' = L&15]
    #pragma unroll
    for (int n = 0; n < 8; ++n)
      xfer[wave][(n + 8 * (lane >> 4)) * 16 + m] = D[n];
    __syncthreads();
    if (lane < 16) {                            // lane owns matrix row = lane
      for (int rp = 15; rp >= 0; --rp) {        // descending r; '>=': smallest r wins ties
        float t = xfer[wave][lane * 16 + rp] + carry;
        int r = kc + rp + 1;
        if (r <= P + 1 && t >= best) { best = t; bestr = r; }
      }
      carry += xfer[wave][lane * 16 + 0];       // add this chunk's full sum
    }
    __syncthreads();
  }
  if (lane < 16 && valid) {
    atomicAdd(&hist[c * (CDIM + 2) + bestr], 1);
    smdot[c * CDIM + row] = v * (Pf + 2.0f - 2.0f * (float)bestr);  // sm[j]*c_minus[j]
    smsum[c * CDIM + row] = v;
  }
}

__global__ void k_finalize(const int* __restrict__ pcnt, const int* __restrict__ poff,
                           int* __restrict__ hist, const float* __restrict__ spsort,
                           const float* __restrict__ smdot, const float* __restrict__ smsum,
                           float* __restrict__ clsloss) {
  __shared__ float red[5][256];
  const int c = blockIdx.x;
  const int P = pcnt[c], N = CDIM - P;
  const int tid = threadIdx.x;
  if (P == 0 || N == 0) { if (tid == 0) clsloss[c] = 0.0f; return; }
  const float Pf = (float)P, Nf = (float)N;
  int* h = hist + c * (CDIM + 2);
  if (tid == 0) {  // in-place cumsum of histogram (exact integer)
    int run = 0;
    for (int t = 0; t <= P + 1; ++t) { run += h[t]; h[t] = run; }
  }
  __syncthreads();
  float aS = 0.f, bS = 0.f, spS = 0.f, dS = 0.f, sS = 0.f;
  const float* sp = spsort + poff[c];
  for (int i = tid; i < P; i += 256) {
    float rp = 1.0f + (float)h[i + 1];          // r_plus[i]
    aS += ((float)i + 1.0f) / ((float)i + rp);
    float s = sp[i];
    bS += s * (Nf + 2.0f - 2.0f * rp);          // dot(sp, c_plus)
    spS += s;
  }
  for (int i = tid; i < CDIM; i += 256) {
    dS += smdot[c * CDIM + i];
    sS += smsum[c * CDIM + i];
  }
  red[0][tid] = aS; red[1][tid] = bS; red[2][tid] = spS; red[3][tid] = dS; red[4][tid] = sS;
  __syncthreads();
  for (int st = 128; st > 0; st >>= 1) {
    if (tid < st)
      for (int q = 0; q < 5; ++q) red[q][tid] += red[q][tid + st];
    __syncthreads();
  }
  if (tid == 0) {
    float DeltaR = 1.0f - red[0][0] / Pf;
    float FR = red[1][0] + red[3][0];
    float FRs = Nf * red[2][0] - Pf * red[4][0];
    clsloss[c] = DeltaR + (FR - FRs) / (Pf * Nf);
  }
}

__global__ void k_total(const float* __restrict__ clsloss, float* __restrict__ out) {
  if (blockIdx.x == 0 && threadIdx.x == 0) {
    float t = 0.0f;
    for (int c = 0; c < NCLS; ++c) t += clsloss[c];
    out[0] = t / (float)NCLS;
  }
}

extern "C" void kernel_launch(void* const* d_in, const int* in_sizes, int n_in,
                              void* d_out, int out_size, void* d_ws, size_t ws_size,
                              hipStream_t stream) {
  const float* scores = (const float*)d_in[0];
  const int* labels = (const int*)d_in[1];
  if (ws_size < WS_TOTAL) return;  // need ~9.3 MB scratch
  char* w = (char*)d_ws;
  float* colsc = (float*)(w + OFF_COLSC);
  unsigned* key32 = (unsigned*)(w + OFF_KEY32);
  unsigned long long* nk = (unsigned long long*)(w + OFF_NK);
  int* negrank = (int*)(w + OFF_NEGRANK);
  float* tblc = (float*)(w + OFF_TBL);
  int* hist = (int*)(w + OFF_HIST);
  float* smdot = (float*)(w + OFF_SMDOT);
  float* smsum = (float*)(w + OFF_SMSUM);
  unsigned long long* posdk = (unsigned long long*)(w + OFF_POSDK);
  float* posval = (float*)(w + OFF_POSVAL);
  float* spsort = (float*)(w + OFF_SPSORT);
  int* pcnt = (int*)(w + OFF_PCNT);
  int* poff = (int*)(w + OFF_POFF);
  int* pfill = (int*)(w + OFF_PFILL);
  float* clsloss = (float*)(w + OFF_CLOSS);

  hipMemsetAsync(d_ws, 0, WS_TOTAL, stream);
  k_build<<<(NCLS * CDIM) / 256, 256, 0, stream>>>(scores, labels, colsc, key32, nk);
  k_count<<<CDIM / 256, 256, 0, stream>>>(labels, pcnt);
  k_prefix<<<1, 32, 0, stream>>>(pcnt, poff);
  k_scatter<<<CDIM / 256, 256, 0, stream>>>(scores, labels, key32, poff, pfill, posdk, posval);
  k_tbl<<<(NCLS * (CDIM + 2) + 255) / 256, 256, 0, stream>>>(pcnt, tblc);
  k_rankpos<<<NCLS, 256, 0, stream>>>(pcnt, poff, posdk, posval, spsort);
  k_negrank<<<dim3(CDIM / 256, NCLS), 256, 0, stream>>>(nk, negrank);
  k_scan_wmma<<<dim3(CDIM / 128, NCLS), 256, 0, stream>>>(colsc, negrank, tblc, spsort,
                                                          pcnt, poff, hist, smdot, smsum);
  k_finalize<<<NCLS, 256, 0, stream>>>(pcnt, poff, hist, spsort, smdot, smsum, clsloss);
  k_total<<<1, 32, 0, stream>>>(clsloss, (float*)d_out);
}